// PhysicalForward_9182640079300
// MI455X (gfx1250) — compile-verified
//
#include <hip/hip_runtime.h>
#include <hip/hip_bf16.h>

typedef float v2f __attribute__((ext_vector_type(2)));
typedef float v8f __attribute__((ext_vector_type(8)));

#define IMG 1024
#define IMGM1 1023.0f
#define NPIX (IMG * IMG)            // 1048576
#define KPSF 21
#define PAD 10
#define PATCH_ROWS 84               // 64 + 2*PAD
#define PATCH_STRIDE 92             // dwords; conflict-free (92 mod 64 = 28, period 16)
#define PATCH_FLOATS (PATCH_ROWS * PATCH_STRIDE)      // 7728
#define NMAT (KPSF * 9)             // 189 Toeplitz B matrices
#define BTAB_FLOATS (NMAT * 64)     // 12096
#define SMEM_FLOATS (PATCH_FLOATS + BTAB_FLOATS + 441 + 1)

// ---------------------------------------------------------------------------
// Kernel 0: scalar parameter setup (softplus/tanh once, not 16.7M times).
// prm = [x0, y0, b, rc*rc + 1e-12, sub0, sub1]
// ---------------------------------------------------------------------------
__global__ void params_kernel(const float* x0, const float* y0,
                              const float* rb, const float* rc,
                              const float* sub, float* prm)
{
    prm[0] = x0[0];
    prm[1] = y0[0];
    prm[2] = log1pf(expf(rb[0])) + 1e-8f;            // softplus + EPS_POS
    float r = log1pf(expf(rc[0])) + 1e-8f;
    prm[3] = r * r + 1e-12f;                         // rc^2 + EPS_R2 (folded)
    prm[4] = 0.25f * tanhf(sub[0]);
    prm[5] = 0.25f * tanhf(sub[1]);
}

// ---------------------------------------------------------------------------
// Kernel 1: warp (radial displacement + bilinear gather).
// 4 j-adjacent pixels per thread -> b128 coalesced stores in the reference's
// transposed (B, W, H) layout: wimg[b][i(x)][j(y)].
// ---------------------------------------------------------------------------
__global__ __launch_bounds__(256) void warp_kernel(
    const float* __restrict__ src, const float* __restrict__ prm,
    float* __restrict__ wimg)
{
    unsigned idx = blockIdx.x * 256u + threadIdx.x;
    int b        = idx >> 18;            // 16 images * 1024 i * 256 j-quads
    unsigned rem = idx & 0x3FFFFu;
    int i        = rem >> 8;             // x index
    int j0       = (rem & 255u) << 2;    // y index (quad base)

    float x0 = prm[0], y0 = prm[1], bb = prm[2], rc2 = prm[3];
    float s0 = prm[4], s1 = prm[5];

    const float step = 2.0f / IMGM1;
    float x   = -1.0f + step * (float)i;
    float dx  = x - x0;
    float dx2 = dx * dx;
    const float* sp = src + (size_t)b * NPIX;

    float4 res;
    float* rp = &res.x;
    #pragma unroll
    for (int s = 0; s < 4; ++s) {
        float y  = -1.0f + step * (float)(j0 + s);
        float dy = y - y0;
        float dn = sqrtf(dx2 + dy * dy + rc2);
        float t  = bb / dn;
        float gx = x - dx * t + s0;
        float gy = y - dy * t + s1;
        float ix = (gx + 1.0f) * (0.5f * IMGM1);
        float iy = (gy + 1.0f) * (0.5f * IMGM1);
        float ix0f = floorf(ix), iy0f = floorf(iy);
        float wx1 = ix - ix0f,  wy1 = iy - iy0f;
        int ix0 = (int)ix0f,    iy0 = (int)iy0f;
        float acc = 0.0f;
        #pragma unroll
        for (int cc = 0; cc < 4; ++cc) {
            int yi = iy0 + (cc >> 1);
            int xi = ix0 + (cc & 1);
            float w = ((cc >> 1) ? wy1 : 1.0f - wy1) *
                      ((cc & 1)  ? wx1 : 1.0f - wx1);
            bool valid = ((unsigned)yi < (unsigned)IMG) &&
                         ((unsigned)xi < (unsigned)IMG);
            int yc = min(max(yi, 0), IMG - 1);
            int xc = min(max(xi, 0), IMG - 1);
            acc += sp[yc * IMG + xc] * (valid ? w : 0.0f);
        }
        rp[s] = acc;
    }
    *(float4*)(wimg + (size_t)b * NPIX + (size_t)i * IMG + j0) = res;
}

// ---------------------------------------------------------------------------
// Kernel 2: 21x21 cross-correlation as Toeplitz WMMA-GEMM.
// WG (256 thr = 8 waves) -> 64x64 output tile:
//  - patch staged via global_load_async_to_lds_b128 (ASYNCcnt), overlapped
//    with PSF normalization + B-table build
//  - 189 B matrices precomputed in LDS in WMMA register layout
//  - each wave: 2 tiles sharing B; inner iter = 3x ds_load_b64 + 2x wmma
// ---------------------------------------------------------------------------
__global__ __launch_bounds__(256) void conv_wmma_kernel(
    const float* __restrict__ wimg,
    const float* __restrict__ raw_psf,
    float* __restrict__ out)
{
    extern __shared__ float smem[];
    float* patch = smem;                         // PATCH_FLOATS
    float* Btab  = smem + PATCH_FLOATS;          // BTAB_FLOATS (16B aligned: 7728*4)
    float* psfs  = Btab + BTAB_FLOATS;           // 441
    float* ssum  = psfs + 441;                   // 1

    const int tid = threadIdx.x;
    const int bj  = blockIdx.x;                  // j tile (fast output dim)
    const int bi  = blockIdx.y;                  // i tile
    const int b   = blockIdx.z;                  // batch

    const int gi0  = bi * 64 - PAD;              // first patch row (global i)
    const int gj0p = bj * 64 - 12;               // first patch col, 16B aligned

    if (tid == 0) ssum[0] = 0.0f;
    // Boundary WGs: pre-zero patch so OOB b128 groups (EXEC-masked off) read 0.
    const bool boundary = (gi0 < 0) | (gi0 + PATCH_ROWS - 1 > IMG - 1) |
                          (gj0p < 0) | (gj0p + 87 > IMG - 1);
    if (boundary) {
        for (int t = tid; t < PATCH_FLOATS; t += 256) patch[t] = 0.0f;
    }
    __syncthreads();

    // ---- Phase 1: fire async DMA of the patch (22 b128 groups x 84 rows) ----
    {
        const float* wb = wimg + (size_t)b * NPIX;
        const unsigned patch_off = (unsigned)(unsigned long long)(uintptr_t)patch;
        for (int t = tid; t < PATCH_ROWS * 22; t += 256) {
            int r  = t / 22;
            int m4 = (t - r * 22) << 2;          // col group base (mult of 4)
            int gi = gi0 + r;
            int gj = gj0p + m4;
            // group is entirely in- or out-of-bounds (edges are mult of 4)
            if ((unsigned)gi < (unsigned)IMG && (unsigned)gj < (unsigned)IMG) {
                const float* g = wb + gi * IMG + gj;
                unsigned l = patch_off + (unsigned)((r * PATCH_STRIDE + m4) * 4);
                asm volatile("global_load_async_to_lds_b128 %0, %1, off"
                             :: "v"(l), "v"(g) : "memory");
            }
        }
    }

    // ---- Phase 1b (overlapped): ReLU PSF + block-wide sum ----
    float part = 0.0f;
    for (int t = tid; t < KPSF * KPSF; t += 256) {
        float v = fmaxf(raw_psf[t], 0.0f);
        psfs[t] = v;
        part += v;
    }
    atomicAdd(ssum, part);
    __syncthreads();

    // ---- Phase 2 (overlapped with DMA): build B matrices in WMMA layout ----
    // Btab[m*64 + lane*2 + v] = psf_n[p][4c + koff + v - n], m = p*9 + c
    {
        float scale = 1.0f / fmaxf(ssum[0], 1e-12f);
        for (int e = tid; e < BTAB_FLOATS; e += 256) {
            int m  = e >> 6;
            int s  = e & 63;
            int ln = s >> 1;
            int v  = s & 1;
            int p  = m / 9;
            int c  = m - 9 * p;
            int n  = ln & 15;
            int ko = (ln >> 4) << 1;
            int d  = 4 * c + ko + v - n;
            float val = ((unsigned)d <= 20u) ? psfs[p * KPSF + d] * scale : 0.0f;
            Btab[e] = val;
        }
    }

    // ---- Phase 3: drain DMA, make patch visible to all waves ----
    asm volatile("s_wait_asynccnt 0" ::: "memory");
    __syncthreads();

    // ---- Phase 4: WMMA compute. Wave wv owns tiles wv and wv+8 (same tj). ----
    const int lane = tid & 31;
    const int wv   = tid >> 5;                 // 0..7
    const int n    = lane & 15;                // N col of D / M row of A
    const int koff = (lane >> 4) << 1;         // A/B: K pair {0,1} or {2,3}
    const int mh   = (lane >> 4) * 8;          // C/D rows r / r+8

    const int ti0 = wv >> 2;                   // 0..1 ; tile1 adds +2
    const int tj0 = wv & 3;
    // A col = tj*16 + k + 2 (patch origin shifted by -12 instead of -10)
    const int acol = tj0 * 16 + koff + 2;

    v8f acc0 = {0.f, 0.f, 0.f, 0.f, 0.f, 0.f, 0.f, 0.f};
    v8f acc1 = {0.f, 0.f, 0.f, 0.f, 0.f, 0.f, 0.f, 0.f};

    for (int p = 0; p < KPSF; ++p) {
        const int arow0 = ti0 * 16 + n + p;
        const float* ap0 = &patch[arow0 * PATCH_STRIDE + acol];
        const float* ap1 = ap0 + 32 * PATCH_STRIDE;     // tile1: ti0+2
        const float* bp  = &Btab[(p * 9) * 64 + lane * 2];
        #pragma unroll
        for (int c = 0; c < 9; ++c) {
            v2f a0 = *(const v2f*)(ap0 + 4 * c);        // ds_load_b64
            v2f a1 = *(const v2f*)(ap1 + 4 * c);
            v2f bm = *(const v2f*)(bp + c * 64);        // linear, conflict-free
            acc0 = __builtin_amdgcn_wmma_f32_16x16x4_f32(
                false, a0, false, bm, (short)0, acc0, false, false);
            acc1 = __builtin_amdgcn_wmma_f32_16x16x4_f32(
                false, a1, false, bm, (short)0, acc1, false, false);
        }
    }

    // ---- Store: VGPR r holds rows I0+mh+r, 16 contiguous cols per half-wave ----
    const int I00 = bi * 64 + ti0 * 16;
    const int J0  = bj * 64 + tj0 * 16;
    float* ob0 = out + (size_t)b * NPIX + (size_t)(I00 + mh) * IMG + (J0 + n);
    float* ob1 = ob0 + 32 * IMG;               // tile1: +2 tile rows
    #pragma unroll
    for (int r = 0; r < 8; ++r) {
        ob0[r * IMG] = acc0[r];
        ob1[r * IMG] = acc1[r];
    }
}

// ---------------------------------------------------------------------------
extern "C" void kernel_launch(void* const* d_in, const int* in_sizes, int n_in,
                              void* d_out, int out_size, void* d_ws, size_t ws_size,
                              hipStream_t stream) {
    const float* src     = (const float*)d_in[0]; // (16,1,1024,1024)
    const float* raw_psf = (const float*)d_in[1]; // (1,1,21,21)
    const float* x0      = (const float*)d_in[2];
    const float* y0      = (const float*)d_in[3];
    const float* raw_b   = (const float*)d_in[4];
    const float* raw_rc  = (const float*)d_in[5];
    const float* sub     = (const float*)d_in[6]; // (2,)
    float* out  = (float*)d_out;
    float* wimg = (float*)d_ws;                   // 64 MB warped image
    float* prm  = wimg + (size_t)16 * NPIX;       // +24 B params

    params_kernel<<<1, 1, 0, stream>>>(x0, y0, raw_b, raw_rc, sub, prm);

    // 16 images * 1024 i * 256 j-quads / 256 threads
    warp_kernel<<<16384, 256, 0, stream>>>(src, prm, wimg);

    conv_wmma_kernel<<<dim3(IMG / 64, IMG / 64, 16), 256,
                       SMEM_FLOATS * sizeof(float), stream>>>(wimg, raw_psf, out);
}